// LinearEncoder_85907935854600
// MI455X (gfx1250) — compile-verified
//
#include <hip/hip_runtime.h>
#include <hip/hip_bf16.h>

typedef __attribute__((ext_vector_type(2))) float v2f;
typedef __attribute__((ext_vector_type(8))) float v8f;

#define IN_CH 128
#define OUT_CH 10
#define XW_STRIDE 16   // padded row stride for xw so gathers can use b128 loads

// ---------------------------------------------------------------------------
// Kernel 1: init out = 0, deg = 1.0 (self-loop pre-counted)
// ---------------------------------------------------------------------------
__global__ __launch_bounds__(256) void gcn_init(float* __restrict__ out,
                                                float* __restrict__ deg, int n) {
    int idx = blockIdx.x * blockDim.x + threadIdx.x;
    if (idx < n * OUT_CH) out[idx] = 0.0f;
    if (idx < n)          deg[idx] = 1.0f;
}

// ---------------------------------------------------------------------------
// Kernel 2: in-degree over dst (edge_index row 1)
// ---------------------------------------------------------------------------
__global__ __launch_bounds__(256) void gcn_degree(const int* __restrict__ ei,
                                                  float* __restrict__ deg, int E) {
    int e = blockIdx.x * blockDim.x + threadIdx.x;
    if (e >= E) return;
    int d = __builtin_nontemporal_load(ei + E + e);
    atomicAdd(&deg[d], 1.0f);
}

// ---------------------------------------------------------------------------
// Kernel 3: xw = x @ W via V_WMMA_F32_16X16X4_F32 (fp32, matches reference).
// One wave computes one 16-row tile; K=128 in 32 steps of 4.
// W staged in LDS zero-padded to 16 columns (keeps EXEC all-ones for WMMA).
// ---------------------------------------------------------------------------
__global__ __launch_bounds__(256) void gcn_xw_wmma(const float* __restrict__ x,
                                                   const float* __restrict__ W,
                                                   float* __restrict__ xw, int n) {
    __shared__ float Wl[IN_CH * 16];
    for (int i = threadIdx.x; i < IN_CH * 16; i += blockDim.x) {
        int kk = i >> 4, nn = i & 15;
        Wl[i] = (nn < OUT_CH) ? W[kk * OUT_CH + nn] : 0.0f;
    }
    __syncthreads();

    const int lane = threadIdx.x & 31;
    const int wave = threadIdx.x >> 5;
    const int tile = blockIdx.x * (blockDim.x >> 5) + wave;
    const int m0   = tile * 16;
    if (m0 >= n) return;                 // wave-uniform branch

    const int half = lane >> 4;          // 0: lanes 0-15, 1: lanes 16-31
    const int l16  = lane & 15;

    v8f acc = {};
    const float* xrow = x + (size_t)(m0 + l16) * IN_CH;
    #pragma unroll 4
    for (int k = 0; k < IN_CH; k += 4) {
        const int ka = k + half * 2;     // this half's K pair
        v2f a;                           // A 16x4: M=l16, K=ka..ka+1
        a.x = xrow[ka + 0];
        a.y = xrow[ka + 1];
        v2f b;                           // B 4x16: N=l16, K=ka..ka+1
        b.x = Wl[(ka + 0) * 16 + l16];
        b.y = Wl[(ka + 1) * 16 + l16];
        acc = __builtin_amdgcn_wmma_f32_16x16x4_f32(
                  /*neg_a=*/false, a, /*neg_b=*/false, b,
                  /*c_mod=*/(short)0, acc, /*reuse_a=*/false, /*reuse_b=*/false);
    }

    // C/D layout: VGPR r -> M = r + 8*half, N = l16
    float* orow = xw + (size_t)m0 * XW_STRIDE;
    #pragma unroll
    for (int r = 0; r < 8; ++r)
        orow[(size_t)(r + 8 * half) * XW_STRIDE + l16] = acc[r];
}

// ---------------------------------------------------------------------------
// Kernel 4: dinv = rsqrt(deg)   (deg >= 1 always due to self-loop)
// ---------------------------------------------------------------------------
__global__ __launch_bounds__(256) void gcn_dinv(const float* __restrict__ deg,
                                                float* __restrict__ dinv, int n) {
    int i = blockIdx.x * blockDim.x + threadIdx.x;
    if (i < n) dinv[i] = rsqrtf(deg[i]);
}

// ---------------------------------------------------------------------------
// Kernel 5: edge scatter: out[dst] += dinv[src]*dinv[dst] * xw[src]
// One thread per edge; xw row gathered with 2x b128 + 1x b64 (stride 64B).
// ---------------------------------------------------------------------------
__global__ __launch_bounds__(256) void gcn_scatter(const int* __restrict__ ei,
                                                   const float* __restrict__ dinv,
                                                   const float* __restrict__ xw,
                                                   float* __restrict__ out, int E) {
    int e = blockIdx.x * blockDim.x + threadIdx.x;
    if (e >= E) return;
    int s = __builtin_nontemporal_load(ei + e);
    int d = __builtin_nontemporal_load(ei + E + e);
    float nrm = dinv[s] * dinv[d];

    const float4* xr = (const float4*)(xw + (size_t)s * XW_STRIDE);
    float4 v0 = xr[0];
    float4 v1 = xr[1];
    float2 v2 = *(const float2*)(xw + (size_t)s * XW_STRIDE + 8);

    float* o = out + (size_t)d * OUT_CH;
    atomicAdd(o + 0, nrm * v0.x);
    atomicAdd(o + 1, nrm * v0.y);
    atomicAdd(o + 2, nrm * v0.z);
    atomicAdd(o + 3, nrm * v0.w);
    atomicAdd(o + 4, nrm * v1.x);
    atomicAdd(o + 5, nrm * v1.y);
    atomicAdd(o + 6, nrm * v1.z);
    atomicAdd(o + 7, nrm * v1.w);
    atomicAdd(o + 8, nrm * v2.x);
    atomicAdd(o + 9, nrm * v2.y);
}

// ---------------------------------------------------------------------------
// Kernel 6: self-loop term + bias: out[i,c] += dinv[i]^2 * xw[i,c] + b[c]
// ---------------------------------------------------------------------------
__global__ __launch_bounds__(256) void gcn_final(const float* __restrict__ xw,
                                                 const float* __restrict__ dinv,
                                                 const float* __restrict__ b,
                                                 float* __restrict__ out, int n) {
    int idx = blockIdx.x * blockDim.x + threadIdx.x;
    if (idx >= n * OUT_CH) return;
    int i = idx / OUT_CH;
    int c = idx - i * OUT_CH;
    float di = dinv[i];
    out[idx] += di * di * xw[(size_t)i * XW_STRIDE + c] + b[c];
}

// ---------------------------------------------------------------------------
extern "C" void kernel_launch(void* const* d_in, const int* in_sizes, int n_in,
                              void* d_out, int out_size, void* d_ws, size_t ws_size,
                              hipStream_t stream) {
    const float* x  = (const float*)d_in[0];
    const int*   ei = (const int*)d_in[1];     // [2, E] flattened
    const float* W  = (const float*)d_in[2];
    const float* b  = (const float*)d_in[3];
    float* out = (float*)d_out;

    const int n = in_sizes[0] / IN_CH;         // 100000
    const int E = in_sizes[1] / 2;             // 3200000

    // workspace carve (aligned)
    int n_pad = ((n + 255) / 256) * 256;
    float* deg  = (float*)d_ws;                 // n floats
    float* dinv = deg + n_pad;                  // n floats
    float* xw   = dinv + n_pad;                 // n * XW_STRIDE floats

    const int B = 256;
    int g_init  = (n * OUT_CH + B - 1) / B;
    int g_edge  = (E + B - 1) / B;
    int tiles   = (n + 15) / 16;
    int g_gemm  = (tiles + (B / 32) - 1) / (B / 32);
    int g_node  = (n + B - 1) / B;
    int g_out   = (n * OUT_CH + B - 1) / B;

    gcn_init   <<<g_init, B, 0, stream>>>(out, deg, n);
    gcn_degree <<<g_edge, B, 0, stream>>>(ei, deg, E);
    gcn_xw_wmma<<<g_gemm, B, 0, stream>>>(x, W, xw, n);
    gcn_dinv   <<<g_node, B, 0, stream>>>(deg, dinv, n);
    gcn_scatter<<<g_edge, B, 0, stream>>>(ei, dinv, xw, out, E);
    gcn_final  <<<g_out,  B, 0, stream>>>(xw, dinv, b, out, n);
}